// GATWithEdgeEmbedding_62251255988694
// MI455X (gfx1250) — compile-verified
//
#include <hip/hip_runtime.h>
#include <cstdint>
#include <cstddef>

#define NEG_SLOPE 0.2f

typedef __attribute__((ext_vector_type(2))) float v2f;
typedef __attribute__((ext_vector_type(8))) float v8f;

// ---------- helpers ----------
__device__ __forceinline__ float lrelu(float x) { return x > 0.f ? x : NEG_SLOPE * x; }

// monotone key: float total order -> unsigned total order
__device__ __forceinline__ unsigned fkey(float f) {
  unsigned u = __float_as_uint(f);
  return (u & 0x80000000u) ? ~u : (u | 0x80000000u);
}

// ---------- fills ----------
__global__ void fill_f32(float* p, size_t n, float v) {
  size_t i = (size_t)blockIdx.x * blockDim.x + threadIdx.x;
  size_t st = (size_t)gridDim.x * blockDim.x;
  for (; i < n; i += st) p[i] = v;
}
__global__ void fill_u32(unsigned* p, size_t n, unsigned v) {
  size_t i = (size_t)blockIdx.x * blockDim.x + threadIdx.x;
  size_t st = (size_t)gridDim.x * blockDim.x;
  for (; i < n; i += st) p[i] = v;
}

// ---------- generic fp32 WMMA GEMM: C[M,N] = A[M,K] @ B[K,N], row-major ----------
// block = 256 threads (8 waves). Block tile: 16 rows x 128 cols (wave = one 16x16 tile).
// A tile staged in LDS (dynamic smem = 16*K*4 bytes), shared by all 8 waves.
__global__ void wmma_gemm_f32(const float* __restrict__ A, const float* __restrict__ B,
                              float* __restrict__ C, int M, int K, int N) {
  extern __shared__ float As[];
  const int rowBase = blockIdx.x * 16;
  const int wave = threadIdx.x >> 5;
  const int lane = threadIdx.x & 31;
  const int colBase = blockIdx.y * 128 + wave * 16;

  for (int idx = threadIdx.x; idx < 16 * K; idx += blockDim.x) {
    int r = idx / K, k = idx - r * K;
    int row = rowBase + r;
    As[idx] = (row < M) ? A[(size_t)row * K + k] : 0.f;
  }
  __syncthreads();

  const int mloc = lane & 15;          // M index within tile (A), N index (B/C)
  const int kb = (lane >> 4) << 1;     // lanes 16-31 hold K+2,K+3
  const int col = colBase + mloc;

  v8f acc = {};
  for (int kk = 0; kk < K; kk += 4) {
    v2f a, b;
    a.x = As[mloc * K + kk + kb];
    a.y = As[mloc * K + kk + kb + 1];
    b.x = B[(size_t)(kk + kb) * N + col];
    b.y = B[(size_t)(kk + kb + 1) * N + col];
    acc = __builtin_amdgcn_wmma_f32_16x16x4_f32(false, a, false, b, (short)0, acc,
                                                false, false);
  }
  const int rOff = (lane >> 4) << 3;   // lanes 16-31 write rows 8..15
  for (int v = 0; v < 8; ++v) {
    int row = rowBase + rOff + v;
    if (row < M) C[(size_t)row * N + col] = acc[v];
  }
}

// ---------- fused: edge_emb = edge_attr @ W_edge + b_edge, scatter-add by dst ----------
// K=64, N=128 fixed. Never materializes the 205MB edge_emb.
__global__ void edge_gemm_scatter(const float* __restrict__ EA, const float* __restrict__ W,
                                  const float* __restrict__ bias, const int* __restrict__ dst,
                                  float* __restrict__ sums, int E) {
  __shared__ float As[16 * 64];
  const int rowBase = blockIdx.x * 16;
  const int wave = threadIdx.x >> 5;
  const int lane = threadIdx.x & 31;
  const int colBase = wave * 16;

  for (int idx = threadIdx.x; idx < 16 * 64; idx += blockDim.x) {
    int r = idx >> 6, k = idx & 63;
    int e = rowBase + r;
    As[idx] = (e < E) ? EA[(size_t)e * 64 + k] : 0.f;
  }
  __syncthreads();

  const int mloc = lane & 15;
  const int kb = (lane >> 4) << 1;
  const int col = colBase + mloc;

  v8f acc = {};
  for (int kk = 0; kk < 64; kk += 4) {
    v2f a, b;
    a.x = As[mloc * 64 + kk + kb];
    a.y = As[mloc * 64 + kk + kb + 1];
    b.x = W[(size_t)(kk + kb) * 128 + col];
    b.y = W[(size_t)(kk + kb + 1) * 128 + col];
    acc = __builtin_amdgcn_wmma_f32_16x16x4_f32(false, a, false, b, (short)0, acc,
                                                false, false);
  }
  const float bv = bias[col];
  const int rOff = (lane >> 4) << 3;
  for (int v = 0; v < 8; ++v) {
    int e = rowBase + rOff + v;
    if (e < E) {
      int d = dst[e];
      atomicAdd(&sums[(size_t)d * 128 + col], acc[v] + bv);
    }
  }
}

__global__ void edge_count(const int* __restrict__ dst, float* __restrict__ cnt, int E) {
  int e = blockIdx.x * blockDim.x + threadIdx.x;
  if (e < E) atomicAdd(&cnt[dst[e]], 1.0f);
}

// x2 = x + sums / max(cnt,1)
__global__ void apply_mean(const float* __restrict__ x, const float* __restrict__ sums,
                           const float* __restrict__ cnt, float* __restrict__ x2, int Nn) {
  size_t i = (size_t)blockIdx.x * blockDim.x + threadIdx.x;
  size_t total = (size_t)Nn * 128;
  if (i < total) {
    int n = (int)(i >> 7);
    x2[i] = x[i] + sums[i] / fmaxf(cnt[n], 1.0f);
  }
}

// as[n,h] = <h1[n,h,:], att_src[h,:]>, ad likewise. one thread per (n,h).
__global__ void attn_coeff(const float* __restrict__ h, const float* __restrict__ att_s,
                           const float* __restrict__ att_d, float* __restrict__ as_,
                           float* __restrict__ ad_, int Nn, int H, int C) {
  int t = blockIdx.x * blockDim.x + threadIdx.x;
  if (t >= Nn * H) return;
  int n = t / H, hh = t - n * H;
  const float* hp = h + (size_t)n * H * C + (size_t)hh * C;
  const float* sp = att_s + (size_t)hh * C;
  const float* dp = att_d + (size_t)hh * C;
  float s = 0.f, d = 0.f;
  for (int c = 0; c < C; ++c) { float v = hp[c]; s += v * sp[c]; d += v * dp[c]; }
  as_[t] = s; ad_[t] = d;
}

// pass 1: segment max via monotone-key atomicMax. edges [0,E)=real, [E,E+Nn)=self loops
__global__ void edge_softmax_max(const int* __restrict__ src, const int* __restrict__ dst,
                                 int E, int Nn, int H, const float* __restrict__ as_,
                                 const float* __restrict__ ad_, unsigned* __restrict__ mkey) {
  int e = blockIdx.x * blockDim.x + threadIdx.x;
  if (e >= E + Nn) return;
  int s = (e < E) ? src[e] : (e - E);
  int d = (e < E) ? dst[e] : (e - E);
  for (int hh = 0; hh < H; ++hh) {
    float ev = lrelu(as_[s * H + hh] + ad_[d * H + hh]);
    atomicMax(&mkey[d * H + hh], fkey(ev));
  }
}

// decode keys back to float in place
__global__ void decode_max(unsigned* mk, size_t n) {
  size_t i = (size_t)blockIdx.x * blockDim.x + threadIdx.x;
  if (i < n) {
    unsigned k = mk[i];
    float f = (k & 0x80000000u) ? __uint_as_float(k & 0x7fffffffu) : __uint_as_float(~k);
    mk[i] = __float_as_uint(f);
  }
}

// pass 2: numerators + segment sum
__global__ void edge_softmax_exp(const int* __restrict__ src, const int* __restrict__ dst,
                                 int E, int Nn, int H, const float* __restrict__ as_,
                                 const float* __restrict__ ad_, const float* __restrict__ m,
                                 float* __restrict__ den, float* __restrict__ alpha) {
  int e = blockIdx.x * blockDim.x + threadIdx.x;
  if (e >= E + Nn) return;
  int s = (e < E) ? src[e] : (e - E);
  int d = (e < E) ? dst[e] : (e - E);
  for (int hh = 0; hh < H; ++hh) {
    float ev = lrelu(as_[s * H + hh] + ad_[d * H + hh]);
    float ex = expf(ev - m[d * H + hh]);
    alpha[(size_t)e * H + hh] = ex;
    atomicAdd(&den[d * H + hh], ex);
  }
}

// pass 3: out[dst, c] += h[src, c] * alpha_norm. one thread per (edge, channel).
__global__ void edge_aggregate(const int* __restrict__ src, const int* __restrict__ dst,
                               int E, int Nn, int H, int C, const float* __restrict__ hbuf,
                               const float* __restrict__ alpha, const float* __restrict__ den,
                               float* __restrict__ out) {
  size_t total = (size_t)(E + Nn) * H * C;
  size_t st = (size_t)gridDim.x * blockDim.x;
  int HC = H * C;
  for (size_t t = (size_t)blockIdx.x * blockDim.x + threadIdx.x; t < total; t += st) {
    int e = (int)(t / HC);
    int c = (int)(t - (size_t)e * HC);
    int hh = c / C;
    int s = (e < E) ? src[e] : (e - E);
    int d = (e < E) ? dst[e] : (e - E);
    float a = alpha[(size_t)e * H + hh] / den[d * H + hh];
    atomicAdd(&out[(size_t)d * HC + c], hbuf[(size_t)s * HC + c] * a);
  }
}

__global__ void bias_relu(float* __restrict__ y, const float* __restrict__ b, size_t total,
                          int stride) {
  size_t i = (size_t)blockIdx.x * blockDim.x + threadIdx.x;
  size_t st = (size_t)gridDim.x * blockDim.x;
  for (; i < total; i += st) y[i] = fmaxf(y[i] + b[i % stride], 0.f);
}

// column sums of y2[N,128] via LDS accumulators, one global atomic per col per block
__global__ void col_sum(const float* __restrict__ y2, float* __restrict__ partial, int Nn) {
  __shared__ float s[128];
  if (threadIdx.x < 128) s[threadIdx.x] = 0.f;
  __syncthreads();
  size_t total = (size_t)Nn * 128;
  size_t st = (size_t)gridDim.x * blockDim.x;
  for (size_t i = (size_t)blockIdx.x * blockDim.x + threadIdx.x; i < total; i += st)
    atomicAdd(&s[i & 127], y2[i]);
  __syncthreads();
  if (threadIdx.x < 128) atomicAdd(&partial[threadIdx.x], s[threadIdx.x]);
}

__global__ void finalize_out(const float* __restrict__ partial, const float* __restrict__ b2,
                             float* __restrict__ out, int Nn) {
  int c = threadIdx.x;
  if (c < 128) out[c] = partial[c] / (float)Nn + b2[c];
}

// ---------- launch ----------
static inline unsigned gs_blocks(size_t n) {
  size_t b = (n + 255) / 256;
  if (b > (size_t)(1u << 20)) b = (size_t)(1u << 20);
  if (b == 0) b = 1;
  return (unsigned)b;
}
static inline unsigned exact_blocks(size_t n) { return (unsigned)((n + 255) / 256); }

extern "C" void kernel_launch(void* const* d_in, const int* in_sizes, int n_in,
                              void* d_out, int out_size, void* d_ws, size_t ws_size,
                              hipStream_t stream) {
  (void)n_in; (void)out_size; (void)ws_size;
  const int Nn = in_sizes[0] / 128;   // 50000 nodes
  const int E  = in_sizes[1] / 2;     // 400000 edges

  const float* x      = (const float*)d_in[0];
  const int*   ei     = (const int*)d_in[1];
  const float* eattr  = (const float*)d_in[2];
  const float* W_edge = (const float*)d_in[3];
  const float* b_edge = (const float*)d_in[4];
  const float* W1     = (const float*)d_in[5];
  const float* att_s1 = (const float*)d_in[6];
  const float* att_d1 = (const float*)d_in[7];
  const float* b1     = (const float*)d_in[8];
  const float* W2     = (const float*)d_in[9];
  const float* att_s2 = (const float*)d_in[10];
  const float* att_d2 = (const float*)d_in[11];
  const float* b2     = (const float*)d_in[12];
  const int* srcA = ei;
  const int* dstA = ei + E;

  // workspace layout (fp32 words)
  float* ws = (float*)d_ws;
  size_t p = 0;
  float* cnt = ws + p; p += (size_t)Nn;
  float* x2  = ws + p; p += (size_t)Nn * 128;
  float* h1  = ws + p; p += (size_t)Nn * 512;   // region reused by layer 2
  float* y1  = ws + p; p += (size_t)Nn * 512;
  float* as1 = ws + p; p += (size_t)Nn * 8;
  float* ad1 = ws + p; p += (size_t)Nn * 8;
  unsigned* mkey1 = (unsigned*)(ws + p); p += (size_t)Nn * 8;  // max keys, decoded in place
  float* den1   = ws + p; p += (size_t)Nn * 8;
  float* alpha1 = ws + p; p += (size_t)(E + Nn) * 8;
  float* partial = ws + p; p += 128;
  // aliases: sums lives in h1's slot before h1 is produced
  float* sums = h1;
  // layer-2 buffers alias the h1 region after h1 is dead (needs 261N+E <= 512N words)
  float* h2 = h1;
  float* y2 = h1 + (size_t)Nn * 128;
  float* as2 = h1 + (size_t)Nn * 256;
  float* ad2 = as2 + Nn;
  unsigned* mkey2 = (unsigned*)(ad2 + Nn);
  float* den2 = (float*)mkey2 + Nn;
  float* alpha2 = den2 + Nn;

  const unsigned KEY_NEG_INF = 0x007FFFFFu;  // fkey(-inf)
  dim3 blk(256);

  // --- edge embedding + scatter-mean into nodes ---
  fill_f32<<<gs_blocks((size_t)Nn * 128), blk, 0, stream>>>(sums, (size_t)Nn * 128, 0.f);
  fill_f32<<<gs_blocks(Nn), blk, 0, stream>>>(cnt, (size_t)Nn, 0.f);
  edge_gemm_scatter<<<(E + 15) / 16, blk, 0, stream>>>(eattr, W_edge, b_edge, dstA, sums, E);
  edge_count<<<exact_blocks(E), blk, 0, stream>>>(dstA, cnt, E);
  apply_mean<<<exact_blocks((size_t)Nn * 128), blk, 0, stream>>>(x, sums, cnt, x2, Nn);

  // --- GAT layer 1: h1 = x2 @ W1 (WMMA, K=128, N=512) ---
  wmma_gemm_f32<<<dim3((Nn + 15) / 16, 4), blk, 16 * 128 * sizeof(float), stream>>>(
      x2, W1, h1, Nn, 128, 512);
  attn_coeff<<<exact_blocks((size_t)Nn * 8), blk, 0, stream>>>(h1, att_s1, att_d1, as1, ad1,
                                                               Nn, 8, 64);
  fill_u32<<<gs_blocks((size_t)Nn * 8), blk, 0, stream>>>(mkey1, (size_t)Nn * 8, KEY_NEG_INF);
  edge_softmax_max<<<exact_blocks(E + Nn), blk, 0, stream>>>(srcA, dstA, E, Nn, 8, as1, ad1,
                                                             mkey1);
  decode_max<<<exact_blocks((size_t)Nn * 8), blk, 0, stream>>>(mkey1, (size_t)Nn * 8);
  fill_f32<<<gs_blocks((size_t)Nn * 8), blk, 0, stream>>>(den1, (size_t)Nn * 8, 0.f);
  edge_softmax_exp<<<exact_blocks(E + Nn), blk, 0, stream>>>(srcA, dstA, E, Nn, 8, as1, ad1,
                                                             (const float*)mkey1, den1, alpha1);
  fill_f32<<<gs_blocks((size_t)Nn * 512), blk, 0, stream>>>(y1, (size_t)Nn * 512, 0.f);
  edge_aggregate<<<gs_blocks((size_t)(E + Nn) * 512), blk, 0, stream>>>(
      srcA, dstA, E, Nn, 8, 64, h1, alpha1, den1, y1);
  bias_relu<<<gs_blocks((size_t)Nn * 512), blk, 0, stream>>>(y1, b1, (size_t)Nn * 512, 512);

  // --- GAT layer 2: h2 = y1 @ W2 (WMMA, K=512, N=128) ---
  wmma_gemm_f32<<<dim3((Nn + 15) / 16, 1), blk, 16 * 512 * sizeof(float), stream>>>(
      y1, W2, h2, Nn, 512, 128);
  attn_coeff<<<exact_blocks(Nn), blk, 0, stream>>>(h2, att_s2, att_d2, as2, ad2, Nn, 1, 128);
  fill_u32<<<gs_blocks(Nn), blk, 0, stream>>>(mkey2, (size_t)Nn, KEY_NEG_INF);
  edge_softmax_max<<<exact_blocks(E + Nn), blk, 0, stream>>>(srcA, dstA, E, Nn, 1, as2, ad2,
                                                             mkey2);
  decode_max<<<exact_blocks(Nn), blk, 0, stream>>>(mkey2, (size_t)Nn);
  fill_f32<<<gs_blocks(Nn), blk, 0, stream>>>(den2, (size_t)Nn, 0.f);
  edge_softmax_exp<<<exact_blocks(E + Nn), blk, 0, stream>>>(srcA, dstA, E, Nn, 1, as2, ad2,
                                                             (const float*)mkey2, den2, alpha2);
  fill_f32<<<gs_blocks((size_t)Nn * 128), blk, 0, stream>>>(y2, (size_t)Nn * 128, 0.f);
  edge_aggregate<<<gs_blocks((size_t)(E + Nn) * 128), blk, 0, stream>>>(
      srcA, dstA, E, Nn, 1, 128, h2, alpha2, den2, y2);

  // --- final mean over nodes + bias ---
  fill_f32<<<1, blk, 0, stream>>>(partial, (size_t)128, 0.f);
  col_sum<<<1024, blk, 0, stream>>>(y2, partial, Nn);
  finalize_out<<<1, 128, 0, stream>>>(partial, b2, (float*)d_out, Nn);
}